// SpikingConv2D_26104811225508
// MI455X (gfx1250) — compile-verified
//
#include <hip/hip_runtime.h>

// SpikingConv2D forward as a tiny-K GEMM on WMMA f32 16x16x4.
//   out[n*128 + f] = min(1.0, sum_k X[n,k]*W[k,f] + (1 - D_i[0,f]))
//   n = (i*256 + j)*16 + b ; X[n,k] = tj[b, i+k/3, j+k%3] ; W[k,f] = kernel_flat[k*128+f]
//
// Bandwidth-bound: 512MB f32 output (~22us floor at 23.3 TB/s). tj (4.3MB)
// lives in L2 with 9x reuse; output streamed with non-temporal stores.

typedef __attribute__((ext_vector_type(2))) float v2f;
typedef __attribute__((ext_vector_type(8))) float v8f;

#define BATCH   16
#define S_IN    258
#define HV      256
#define FC      128
#define WAVES   4   // waves per block

__global__ __launch_bounds__(WAVES * 32) void spiking_conv2d_wmma(
    const float* __restrict__ tj,     // [16, 258, 258]
    const float* __restrict__ Wk,     // flat 1152 floats, W[k,f] = Wk[k*128 + f]
    const float* __restrict__ Di,     // [9,128]; row 0 used
    float* __restrict__ out)          // [N, 128] flat
{
    const int tid  = threadIdx.x;
    const int lane = tid & 31;
    const int wave = tid >> 5;
    const int half = lane >> 4;   // 0: lanes 0-15, 1: lanes 16-31
    const int col  = lane & 15;   // N-index within a 16-col F tile / M-index for A

    // one wave == one output pixel (16 batch rows x 128 filters)
    const int pix = blockIdx.x * WAVES + wave;      // 0 .. 65535
    const int i   = pix >> 8;                       // output row
    const int j   = pix & 255;                      // output col
    const int rowBase = pix * BATCH;                // first GEMM row of this tile

    // ---- A fragment: 16x9 f32 tile (K padded to 12 -> 3 steps of K=4) ----
    // A 16x4 layout: lanes 0-15 hold M=lane, K={4s,4s+1}; lanes 16-31 hold K={4s+2,4s+3}.
    // Here M == batch index b, so b = col for both lane halves.
    const float* tjb = tj + (size_t)col * (S_IN * S_IN) + (size_t)i * S_IN + j;
    v2f A[3];
#pragma unroll
    for (int s = 0; s < 3; ++s) {
        const int k0 = 4 * s + 2 * half;
        const int k1 = k0 + 1;
        const int c0 = k0 > 8 ? 8 : k0;         // clamp for safe address
        const int c1 = k1 > 8 ? 8 : k1;
        const float a0 = tjb[(c0 / 3) * S_IN + (c0 % 3)];
        const float a1 = tjb[(c1 / 3) * S_IN + (c1 % 3)];
        A[s].x = (k0 <= 8) ? a0 : 0.0f;         // lane-level select, no divergence
        A[s].y = (k1 <= 8) ? a1 : 0.0f;
    }

    // ---- 8 F-tiles of 16 columns each ----
#pragma unroll
    for (int t = 0; t < 8; ++t) {
        const int f0 = t * 16;

        // threshold bias: C[m,n] = 1 - D_i[0, f0+n]; same for every row m,
        // and C/D layout has N = lane&15 in every VGPR -> broadcast per lane.
        const float thr = 1.0f - Di[f0 + col];
        v8f acc;
#pragma unroll
        for (int v = 0; v < 8; ++v) acc[v] = thr;

        // B 4x16 layout mirrors A: VGPR0 = K={4s (lanes 0-15), 4s+2 (16-31)},
        // VGPR1 = K+1; N = lane&15.
#pragma unroll
        for (int s = 0; s < 3; ++s) {
            const int k0 = 4 * s + 2 * half;
            const int k1 = k0 + 1;
            const int c0 = k0 > 8 ? 8 : k0;
            const int c1 = k1 > 8 ? 8 : k1;
            v2f B;
            const float b0 = Wk[c0 * FC + f0 + col];
            const float b1 = Wk[c1 * FC + f0 + col];
            B.x = (k0 <= 8) ? b0 : 0.0f;
            B.y = (k1 <= 8) ? b1 : 0.0f;

            acc = __builtin_amdgcn_wmma_f32_16x16x4_f32(
                /*neg_a=*/false, A[s], /*neg_b=*/false, B,
                /*c_mod=*/(short)0, acc, /*reuse_a=*/false, /*reuse_b=*/false);
        }

        // ---- clamp + streaming store ----
        // C/D: VGPR v -> row M = v + 8*half, column N = col. Half-wave writes
        // 64B contiguous; stream past caches (output is 512MB, one-touch).
        float* orow = out + (size_t)(rowBase + 8 * half) * FC + f0 + col;
#pragma unroll
        for (int v = 0; v < 8; ++v) {
            const float val = fminf(acc[v], 1.0f);
            __builtin_nontemporal_store(val, orow + (size_t)v * FC);
        }
    }
}

extern "C" void kernel_launch(void* const* d_in, const int* in_sizes, int n_in,
                              void* d_out, int out_size, void* d_ws, size_t ws_size,
                              hipStream_t stream) {
    const float* tj  = (const float*)d_in[0];   // [16,258,258]
    const float* Wk  = (const float*)d_in[1];   // 1152 floats
    const float* Di  = (const float*)d_in[2];   // [9,128]
    float*       out = (float*)d_out;           // 134,217,728 floats

    const int pixels = HV * HV;                 // 65536 waves total
    dim3 block(WAVES * 32);
    dim3 grid(pixels / WAVES);                  // 16384 blocks, exact
    spiking_conv2d_wmma<<<grid, block, 0, stream>>>(tj, Wk, Di, out);
    (void)in_sizes; (void)n_in; (void)out_size; (void)d_ws; (void)ws_size;
}